// RTDETRPostProcessor_12661563589125
// MI455X (gfx1250) — compile-verified
//
#include <hip/hip_runtime.h>
#include <math.h>

#define NTH 1024
#define QN 1000
#define CN 80
#define KTOP 300
#define NF4 20000   // 1000*80/4 float4 elements per batch
#define RECCAP 2048

typedef float v2f __attribute__((ext_vector_type(2)));
typedef float v8f __attribute__((ext_vector_type(8)));

__device__ __forceinline__ unsigned f2k(float f) {
  unsigned u = __float_as_uint(f);
  return (u & 0x80000000u) ? ~u : (u | 0x80000000u);
}
__device__ __forceinline__ float k2f(unsigned k) {
  unsigned u = (k & 0x80000000u) ? (k ^ 0x80000000u) : ~k;
  return __uint_as_float(u);
}
__device__ __forceinline__ unsigned umin2(unsigned a, unsigned b) { return a < b ? a : b; }
__device__ __forceinline__ unsigned umax2(unsigned a, unsigned b) { return a > b ? a : b; }

__global__ __launch_bounds__(NTH) void rtdetr_post_kernel(
    const float* __restrict__ logits, const float* __restrict__ boxes,
    const float* __restrict__ osz, float* __restrict__ labels_out,
    float* __restrict__ boxes_out, float* __restrict__ scores_out) {
  const int b = blockIdx.x;
  const int tid = (int)threadIdx.x;
  const int wave = tid >> 5;
  const int lane = tid & 31;
  const float* L = logits + (size_t)b * QN * CN;
  const float4* L4 = (const float4*)L;
  const float* BX = boxes + (size_t)b * QN * 4;

  __shared__ unsigned hist[8][2048];     // 64 KB, wave-privatized
  __shared__ unsigned boxmaxk[1024];     // per-query max key
  __shared__ unsigned selFlag[1024];
  __shared__ unsigned scanA[1024];
  __shared__ int uniq[304];
  __shared__ unsigned jkey[304];
  __shared__ unsigned recT[RECCAP];      // flat idx of prefix-matching entries
  __shared__ unsigned recK[RECCAP];      // their keys
  __shared__ float ldsBoxes[QN * 4];
  __shared__ unsigned sPrefix, sK, sRecCount, sNuniq;

  // ---- async preload of boxes tile into LDS (overlaps the selection scans) ----
  for (int i = tid; i < QN * 4; i += NTH) {
#if __has_builtin(__builtin_amdgcn_global_load_async_to_lds_b32)
    __builtin_amdgcn_global_load_async_to_lds_b32(
        (__attribute__((address_space(1))) int*)(BX + i),
        (__attribute__((address_space(3))) int*)(&ldsBoxes[i]), 0, 0);
#else
    ldsBoxes[i] = BX[i];
#endif
  }

  if (tid == 0) { sPrefix = 0u; sK = KTOP; sRecCount = 0u; }
  boxmaxk[tid] = 0u;
  if (tid < 304) { uniq[tid] = 0; jkey[tid] = 0u; }
  __syncthreads();

  // ---- 3-pass (11+11+10 bit) radix select for the exact 300th-largest key ----
  const int SH[3] = {21, 10, 0};
  const int BB[3] = {11, 11, 10};
  for (int pass = 0; pass < 3; ++pass) {
    const int shift = SH[pass];
    const int binbits = BB[pass];
    const int nb = 1 << binbits;
    for (int i = tid; i < 8 * nb; i += NTH) (&hist[0][0])[i] = 0u;
    __syncthreads();
    const unsigned pfx = sPrefix;
    unsigned* myh = hist[wave & 7];
    for (int t4 = tid; t4 < NF4; t4 += NTH) {
      const float4 v = L4[t4];
      const int base = t4 * 4;
      const unsigned q = (unsigned)base / CN;
      const int c0 = base - (int)q * CN;  // 4-aligned, never crosses a query row
      unsigned kk[4];
      kk[0] = f2k(v.x); kk[1] = f2k(v.y); kk[2] = f2k(v.z); kk[3] = f2k(v.w);
      if (pass == 0) {  // fold per-query max into first scan
        unsigned mk = 0u;
#pragma unroll
        for (int e = 0; e < 4; ++e)
          if (c0 + e != 0) mk = umax2(mk, kk[e]);
        atomicMax(&boxmaxk[q], mk);
      }
#pragma unroll
      for (int e = 0; e < 4; ++e) {
        if (c0 + e == 0) continue;  // class 0 excluded (logits[:,:,1:])
        const unsigned k = kk[e];
        if (pass == 0 || (k >> (shift + binbits)) == pfx) {
          atomicAdd(&myh[(k >> shift) & (unsigned)(nb - 1)], 1u);
          if (pass == 2) {  // record candidates containing all threshold-equal entries
            const unsigned p = atomicAdd(&sRecCount, 1u);
            if (p < RECCAP) { recK[p] = k; recT[p] = (unsigned)(base + e); }
          }
        }
      }
    }
    __syncthreads();
    // reduce privatized copies
    for (int bin = tid; bin < nb; bin += NTH) {
      unsigned s = 0;
#pragma unroll
      for (int cpy = 0; cpy < 8; ++cpy) s += hist[cpy][bin];
      hist[0][bin] = s;
    }
    __syncthreads();
    // parallel descending suffix-sum + boundary pick
    const int half = nb >> 1;
    const unsigned need = sK;
    const unsigned pfxl = sPrefix;
    unsigned h0 = 0, h1 = 0, ps = 0;
    if (tid < half) {
      h0 = hist[0][(nb - 1) - 2 * tid];
      h1 = hist[0][(nb - 2) - 2 * tid];
      ps = h0 + h1;
    }
    scanA[tid] = (tid < half) ? ps : 0u;
    __syncthreads();
    for (int off = 1; off < 1024; off <<= 1) {
      const unsigned cur = scanA[tid];
      const unsigned add = (tid >= off) ? scanA[tid - off] : 0u;
      __syncthreads();
      scanA[tid] = cur + add;
      __syncthreads();
    }
    if (tid < half) {
      const unsigned excl = scanA[tid] - ps;
      const unsigned ge0 = excl + h0;        // cnt(key-bin >= bin0)
      const unsigned ge1 = excl + h0 + h1;   // cnt(key-bin >= bin1)
      const int bin0 = (nb - 1) - 2 * tid;
      if (ge0 >= need && (ge0 - h0) < need) {
        sPrefix = (pfxl << binbits) | (unsigned)bin0;
        sK = need - (ge0 - h0);
      }
      if (ge1 >= need && (ge1 - h1) < need) {
        sPrefix = (pfxl << binbits) | (unsigned)(bin0 - 1);
        sK = need - (ge1 - h1);
      }
    }
    __syncthreads();
  }
  const unsigned Tkey = sPrefix;
  unsigned m = sK;  // #threshold-equal entries to take (smallest flat index first)

  // ---- equals candidates; fallback rescan if the record buffer overflowed ----
  const unsigned recTotal0 = sRecCount;
  __syncthreads();
  bool fellback = false;
  if (recTotal0 > RECCAP) {
    fellback = true;
    if (tid == 0) sRecCount = 0u;
    __syncthreads();
    for (int t4 = tid; t4 < NF4; t4 += NTH) {
      const float4 v = L4[t4];
      const int base = t4 * 4;
      const int c0 = base - (base / CN) * CN;
      const float vv[4] = {v.x, v.y, v.z, v.w};
#pragma unroll
      for (int e = 0; e < 4; ++e) {
        if (c0 + e == 0) continue;
        if (f2k(vv[e]) == Tkey) {
          const unsigned p = atomicAdd(&sRecCount, 1u);
          if (p < RECCAP) recT[p] = (unsigned)(base + e);
        }
      }
    }
    __syncthreads();
  }
  const unsigned nrec = umin2(sRecCount, RECCAP);

  // selFlag directly from per-query max (gt-selected queries)
  selFlag[tid] = (tid < QN && boxmaxk[tid] > Tkey) ? 1u : 0u;
  // invalidate non-equal records; keep flat index for tie-break ordering
  for (int p = tid; p < RECCAP; p += NTH) {
    const bool valid = (p < (int)nrec) && (fellback || recK[p] == Tkey);
    if (!valid) recT[p] = 0xFFFFFFFFu;
  }
  __syncthreads();

  // take the m smallest flat indices among threshold-equal entries
  for (unsigned it = 0; it < m; ++it) {
    scanA[tid] = umin2(recT[tid], recT[tid + 1024]);
    __syncthreads();
    for (int off = 512; off > 0; off >>= 1) {
      if (tid < off) scanA[tid] = umin2(scanA[tid], scanA[tid + off]);
      __syncthreads();
    }
    const unsigned mn = scanA[0];
    __syncthreads();
    if (mn == 0xFFFFFFFFu) break;
    if (tid == 0) selFlag[mn / CN] = 1u;
    if (recT[tid] == mn) recT[tid] = 0xFFFFFFFFu;
    if (recT[tid + 1024] == mn) recT[tid + 1024] = 0xFFFFFFFFu;
    __syncthreads();
  }

  // ---- compact selected queries (ascending) into uniq[], zero padded ----
  const unsigned flag = (tid < QN) ? selFlag[tid] : 0u;
  scanA[tid] = flag;
  __syncthreads();
  for (int off = 1; off < 1024; off <<= 1) {
    const unsigned cur = scanA[tid];
    const unsigned add = (tid >= off) ? scanA[tid - off] : 0u;
    __syncthreads();
    scanA[tid] = cur + add;
    __syncthreads();
  }
  if (tid < QN && flag) {
    const int pos = (int)scanA[tid] - 1;
    uniq[pos] = tid;
    jkey[pos] = boxmaxk[tid];  // == max selected key for this query
  }
  if (tid == 0) sNuniq = scanA[1023];
  __syncthreads();

  // drain the async boxes prefetch before consuming LDS boxes
#if __has_builtin(__builtin_amdgcn_s_wait_asynccnt)
  __builtin_amdgcn_s_wait_asynccnt(0);
#else
  asm volatile("s_wait_asynccnt 0" ::: "memory");
#endif
  __syncthreads();

  const int nuniq = (int)sNuniq;

  // ---- scores: sigmoid of per-box max logit (monotone key inverse) ----
  if (tid < KTOP) {
    float sc = 0.0f;
    if (tid < nuniq) {
      const float lv = k2f(jkey[tid]);
      sc = 1.0f / (1.0f + expf(-lv));
    }
    scores_out[b * KTOP + tid] = sc;
  }

  // ---- labels: softmax over classes 1..79, keep first 4 (one wave per j) ----
  for (int j = wave; j < KTOP; j += 32) {
    const int q = uniq[j];
    const float* lq = L + q * CN;
    const float v0 = lq[1 + lane];
    const float v1 = lq[33 + lane];
    const bool has2 = (65 + lane) < CN;
    const float v2 = has2 ? lq[65 + lane] : -__builtin_inff();
    float mx = fmaxf(fmaxf(v0, v1), v2);
    for (int o = 16; o > 0; o >>= 1) mx = fmaxf(mx, __shfl_xor(mx, o, 32));
    float se = expf(v0 - mx) + expf(v1 - mx) + (has2 ? expf(v2 - mx) : 0.0f);
    for (int o = 16; o > 0; o >>= 1) se += __shfl_xor(se, o, 32);
    if (lane < 4)
      labels_out[(size_t)(b * KTOP + j) * 4 + lane] = expf(v0 - mx) / se;
  }

  // ---- boxes: cxcywh->xyxy + per-batch scaling as one 16x16x4 f32 WMMA per 16 boxes ----
  const float s0 = osz[b * 2 + 0];
  const float s1 = osz[b * 2 + 1];
  for (int tile = wave; tile < (KTOP + 15) / 16; tile += 32) {
    const int j0 = tile * 16;
    const int r = lane & 15;
    const int q = uniq[j0 + r];  // uniq[] sized 304, zero padded
    const float* bq = &ldsBoxes[q * 4];
    // A (16x4): lanes 0-15 hold (K0,K1)=(cx,cy); lanes 16-31 hold (K2,K3)=(w,h)
    v2f a;
    if (lane < 16) { a.x = bq[0]; a.y = bq[1]; }
    else           { a.x = bq[2]; a.y = bq[3]; }
    // B (4x16): lanes 0-15 hold rows (K0,K1); lanes 16-31 hold rows (K2,K3).
    // Columns 0..3 = [s0,0,-s0/2,0; 0,s1,0,-s1/2; s0,0,s0/2,0; 0,s1,0,s1/2]^T
    v2f bm; bm.x = 0.0f; bm.y = 0.0f;
    if (lane < 16) {
      if (lane == 0 || lane == 2) bm.x = s0;
      if (lane == 1 || lane == 3) bm.y = s1;
    } else {
      const int n = lane - 16;
      if (n == 0) bm.x = -0.5f * s0;
      if (n == 2) bm.x =  0.5f * s0;
      if (n == 1) bm.y = -0.5f * s1;
      if (n == 3) bm.y =  0.5f * s1;
    }
    v8f c = {0.f, 0.f, 0.f, 0.f, 0.f, 0.f, 0.f, 0.f};
    v8f d = __builtin_amdgcn_wmma_f32_16x16x4_f32(false, a, false, bm,
                                                  (short)0, c, false, false);
    // D (16x16 f32): VGPR rr, lanes 0-15 -> D[rr][lane]; lanes 16-31 -> D[rr+8][lane-16]
    const int col = lane & 15;
    if (col < 4) {
      const int rbase = j0 + ((lane & 16) ? 8 : 0);
#pragma unroll
      for (int rr = 0; rr < 8; ++rr) {
        const int row = rbase + rr;
        if (row < KTOP)
          boxes_out[(size_t)(b * KTOP + row) * 4 + col] = d[rr];
      }
    }
  }
}

extern "C" void kernel_launch(void* const* d_in, const int* in_sizes, int n_in,
                              void* d_out, int out_size, void* d_ws, size_t ws_size,
                              hipStream_t stream) {
  const float* logits = (const float*)d_in[0];
  const float* boxes = (const float*)d_in[1];
  const float* osz = (const float*)d_in[2];
  const int Bn = in_sizes[2] / 2;  // 256 batches
  float* out = (float*)d_out;
  float* labels_out = out;                                // [B,300,4]
  float* boxes_out = out + (size_t)Bn * KTOP * 4;         // [B,300,4]
  float* scores_out = out + (size_t)Bn * KTOP * 8;        // [B,300]
  rtdetr_post_kernel<<<Bn, NTH, 0, stream>>>(logits, boxes, osz,
                                             labels_out, boxes_out, scores_out);
}